// SpikingMambaBlock_10136122818950
// MI455X (gfx1250) — compile-verified
//
#include <hip/hip_runtime.h>

// ---------------- problem constants ----------------
#define D_MODEL 1024
#define D_INNER 2048
#define D_STATE 16
#define D_CONV  4
#define DT_RANK 64
#define B_SZ    4
#define SEQ_L   2048
#define IN_DIM  (2*D_INNER + 2*D_STATE + DT_RANK)   // 4192
#define NROWS   (B_SZ*SEQ_L)                         // 8192
// compact proj buffer: [x_in (2048) | B (16) | C (16) | dtr (64)] = 2144 cols
#define PCOLS   (D_INNER + 2*D_STATE + DT_RANK)      // 2144
#define COL_B   (D_INNER)                            // 2048
#define COL_C   (D_INNER + D_STATE)                  // 2064
#define COL_DTR (D_INNER + 2*D_STATE)                // 2080

typedef __attribute__((ext_vector_type(2))) float v2f;
typedef __attribute__((ext_vector_type(8))) float v8f;

// =====================================================================
// Kernel 1: in_proj GEMM (fp32 WMMA 16x16x4), weight columns 2048..4191.
//   P[m, c] = sum_k X[m,k] * W[k, 2048+c] + b[2048+c]
// Block = 4 waves; block tile 128(M) x 64(N); wave tile 32x64 (2 M-tiles
// x 4 N-tiles, 8 accumulators). B staged in LDS per 16-deep k-chunk,
// PRE-PAIRED as float2 {B[k][c], B[k+1][c]} so each WMMA B-fragment is a
// single ds_load_b64 straight into an aligned VGPR pair (no repack movs).
// Pair-row stride 48 float2 = 96 dwords == 32 banks (mod 64): lane halves
// hit disjoint 32-aligned bank windows -> conflict-free b64 reads.
// =====================================================================
#define BLK_K   16
#define PAIRSTR 48   // float2 stride per pair-row

__global__ __launch_bounds__(128) void gemm_inproj(
    const float* __restrict__ X, const float* __restrict__ W,
    const float* __restrict__ bias, float* __restrict__ P)
{
    __shared__ v2f Bs[(BLK_K / 2) * PAIRSTR];         // 8 pair-rows x 48 -> 3 KB
    const int tid  = threadIdx.y * 32 + threadIdx.x;
    const int lane = threadIdx.x;
    const int half = lane >> 4, lm = lane & 15;
    const int m0 = blockIdx.y * 128 + threadIdx.y * 32;
    const int n0 = blockIdx.x << 6;

    v8f acc[2][4] = {};
    const float* arow0 = X + (size_t)(m0 + lm) * D_MODEL;
    const float* arow1 = X + (size_t)(m0 + 16 + lm) * D_MODEL;

    for (int kc = 0; kc < D_MODEL; kc += BLK_K) {
        __syncthreads();                              // protect previous chunk reads
        // cooperative fill: 8 pair-rows x 64 cols (512 float2; 4 per thread)
#pragma unroll
        for (int it = 0; it < ((BLK_K / 2) * 64) / 128; ++it) {
            int idx = it * 128 + tid;
            int kp = idx >> 6, c = idx & 63;
            int col = n0 + c;
            int wc = D_INNER + (col < PCOLS ? col : PCOLS - 1);   // clamp OOB cols
            const float* wr = W + (size_t)(kc + kp * 2) * IN_DIM + wc;
            v2f w2; w2.x = wr[0]; w2.y = wr[IN_DIM];
            Bs[kp * PAIRSTR + c] = w2;
        }
        __syncthreads();
#pragma unroll
        for (int kk = 0; kk < BLK_K; kk += 4) {
            const int ka = kk + 2 * half;
            v2f a0 = *(const v2f*)(arow0 + kc + ka);  // A frags: K=ka, ka+1
            v2f a1 = *(const v2f*)(arow1 + kc + ka);
            const v2f* brow = &Bs[((kk >> 1) + half) * PAIRSTR];
#pragma unroll
            for (int j = 0; j < 4; ++j) {
                v2f bm = brow[(j << 4) + lm];         // one ds_load_b64
                acc[0][j] = __builtin_amdgcn_wmma_f32_16x16x4_f32(
                    false, a0, false, bm, (short)0, acc[0][j], false, false);
                acc[1][j] = __builtin_amdgcn_wmma_f32_16x16x4_f32(
                    false, a1, false, bm, (short)0, acc[1][j], false, false);
            }
        }
    }
#pragma unroll
    for (int j = 0; j < 4; ++j) {
        int c = n0 + (j << 4) + lm;
        if (c < PCOLS) {
            float bv = bias[D_INNER + c];
#pragma unroll
            for (int i = 0; i < 2; ++i) {
#pragma unroll
                for (int r = 0; r < 8; ++r) {
                    int row = m0 + i * 16 + r + (half << 3);
                    P[(size_t)row * PCOLS + c] = acc[i][j][r] + bv;
                }
            }
        }
    }
}

// =====================================================================
// Kernel 2: dt GEMM (fp32 WMMA, K=64) + fused softplus
//   DT[m, n] = softplus( sum_k P[m, COL_DTR+k] * Wdt[k,n] + bdt[n] )
// =====================================================================
__global__ __launch_bounds__(128) void gemm_dt(
    const float* __restrict__ P, const float* __restrict__ W,
    const float* __restrict__ bias, float* __restrict__ DT)
{
    const int lane = threadIdx.x;
    const int half = lane >> 4, lm = lane & 15;
    const int m0 = (blockIdx.y * 4 + threadIdx.y) << 4;
    const int n0 = blockIdx.x << 6;
    v8f acc[4] = {};
    const float* arow = P + (size_t)(m0 + lm) * PCOLS + COL_DTR;
#pragma unroll
    for (int k = 0; k < DT_RANK; k += 4) {
        const int ka = k + 2 * half;
        v2f a = *(const v2f*)(arow + ka);
        const float* wr0 = W + (size_t)ka * D_INNER;
        const float* wr1 = wr0 + D_INNER;
#pragma unroll
        for (int j = 0; j < 4; ++j) {
            int c = n0 + (j << 4) + lm;
            v2f bm; bm.x = wr0[c]; bm.y = wr1[c];
            acc[j] = __builtin_amdgcn_wmma_f32_16x16x4_f32(
                false, a, false, bm, (short)0, acc[j], false, false);
        }
    }
#pragma unroll
    for (int j = 0; j < 4; ++j) {
        int c = n0 + (j << 4) + lm;
        float bv = bias[c];
#pragma unroll
        for (int r = 0; r < 8; ++r) {
            int row = m0 + r + (half << 3);
            float v = acc[j][r] + bv;
            float sp = (v > 20.f) ? v : logf(1.f + expf(v));   // softplus
            DT[(size_t)row * D_INNER + c] = sp;
        }
    }
}

// =====================================================================
// Kernel 3: z at last timestep only (4 rows x 2048 cols GEMV)
// =====================================================================
__global__ void zlast_kernel(const float* __restrict__ X, const float* __restrict__ W,
                             const float* __restrict__ bias, float* __restrict__ zl)
{
    int i = blockIdx.x * blockDim.x + threadIdx.x;   // 0..8191
    int b = i >> 11, d = i & (D_INNER - 1);
    const float* xr = X + (size_t)(b * SEQ_L + SEQ_L - 1) * D_MODEL;
    float s = bias[d];
    for (int k = 0; k < D_MODEL; ++k)
        s += xr[k] * W[(size_t)k * IN_DIM + d];      // coalesced over d
    zl[i] = s;
}

// =====================================================================
// Kernel 4: selective scan, fused depthwise conv + SiLU.
// One lane per (b, d, n): h_n recurrence; 16-lane shfl_xor reduce for y.
// =====================================================================
__global__ __launch_bounds__(256) void scan_kernel(
    const float* __restrict__ P, const float* __restrict__ DT,
    const float* __restrict__ convw, const float* __restrict__ convb,
    const float* __restrict__ Alog,
    float* __restrict__ ylast, float* __restrict__ xclast)
{
    const int tid = threadIdx.x;
    const int n  = tid & 15;
    const int dl = tid >> 4;
    const int d  = blockIdx.x * 16 + dl;
    const int b  = blockIdx.y;
    const float Aval = -expf(Alog[d * D_STATE + n]);
    const float w0 = convw[d*4+0], w1 = convw[d*4+1];
    const float w2 = convw[d*4+2], w3 = convw[d*4+3];
    const float cb = convb[d];
    float h = 0.f, x1 = 0.f, x2 = 0.f, x3 = 0.f, xcs = 0.f;
    const float* prow  = P  + (size_t)b * SEQ_L * PCOLS;
    const float* dtrow = DT + (size_t)b * SEQ_L * D_INNER;
    for (int t = 0; t < SEQ_L; ++t) {
        const float* pr = prow + (size_t)t * PCOLS;
        float xt = pr[d];                                     // x_in
        float xc = w0*x3 + w1*x2 + w2*x1 + w3*xt + cb;        // causal depthwise conv
        xcs = xc / (1.f + expf(-xc));                         // SiLU
        float dtv = dtrow[(size_t)t * D_INNER + d];
        float Bn  = pr[COL_B + n];
        h = expf(Aval * dtv) * h + (dtv * xcs) * Bn;          // selective SSM step
        x3 = x2; x2 = x1; x1 = xt;
    }
    float Cn = prow[(size_t)(SEQ_L - 1) * PCOLS + COL_C + n];
    float p = h * Cn;
    p += __shfl_xor(p, 1, 32);
    p += __shfl_xor(p, 2, 32);
    p += __shfl_xor(p, 4, 32);
    p += __shfl_xor(p, 8, 32);                                // sum over 16 states
    if (n == 0) {
        ylast[b * D_INNER + d]  = p;
        xclast[b * D_INNER + d] = xcs;                        // x_conv at t=L-1
    }
}

// =====================================================================
// Kernel 5: gate + layernorm of the final feature vector (per batch)
// =====================================================================
__global__ __launch_bounds__(256) void finalize_ln(
    const float* __restrict__ ylast, const float* __restrict__ xclast,
    const float* __restrict__ zlast, const float* __restrict__ Dp,
    const float* __restrict__ g, const float* __restrict__ be,
    float* __restrict__ fm)
{
    __shared__ float ssum[256], ssq[256];
    const int b = blockIdx.x, tid = threadIdx.x;
    float s = 0.f, sq = 0.f;
    for (int d = tid; d < D_INNER; d += 256) {
        float y = ylast[b*D_INNER + d] + xclast[b*D_INNER + d] * Dp[d];
        float z = zlast[b*D_INNER + d];
        y *= z / (1.f + expf(-z));                 // y * silu(z)
        fm[b*D_INNER + d] = y;
        s += y; sq += y * y;
    }
    ssum[tid] = s; ssq[tid] = sq; __syncthreads();
    for (int o = 128; o > 0; o >>= 1) {
        if (tid < o) { ssum[tid] += ssum[tid+o]; ssq[tid] += ssq[tid+o]; }
        __syncthreads();
    }
    float mu  = ssum[0] / (float)D_INNER;
    float var = ssq[0] / (float)D_INNER - mu * mu;
    float rs  = rsqrtf(var + 1e-5f);
    for (int d = tid; d < D_INNER; d += 256) {
        float y = fm[b*D_INNER + d];               // same thread wrote these indices
        fm[b*D_INNER + d] = (y - mu) * rs * g[d] + be[d];
    }
}

// =====================================================================
// Kernel 6: out projection (4 x 2048) @ (2048 x 1024)
// =====================================================================
__global__ void outproj_kernel(const float* __restrict__ fm, const float* __restrict__ W,
                               const float* __restrict__ bias, float* __restrict__ out)
{
    int i = blockIdx.x * blockDim.x + threadIdx.x;  // 0..4095
    int b = i >> 10, m = i & (D_MODEL - 1);
    float s = bias[m];
    const float* f = fm + (size_t)b * D_INNER;
    for (int d = 0; d < D_INNER; ++d)
        s += f[d] * W[(size_t)d * D_MODEL + m];     // coalesced over m
    out[i] = s;
}

// =====================================================================
extern "C" void kernel_launch(void* const* d_in, const int* in_sizes, int n_in,
                              void* d_out, int out_size, void* d_ws, size_t ws_size,
                              hipStream_t stream)
{
    const float* x          = (const float*)d_in[0];
    const float* in_proj_w  = (const float*)d_in[1];
    const float* in_proj_b  = (const float*)d_in[2];
    const float* conv_w     = (const float*)d_in[3];
    const float* conv_b     = (const float*)d_in[4];
    const float* dt_proj_w  = (const float*)d_in[5];
    const float* dt_proj_b  = (const float*)d_in[6];
    const float* A_log      = (const float*)d_in[7];
    const float* D_param    = (const float*)d_in[8];
    const float* out_proj_w = (const float*)d_in[9];
    const float* out_proj_b = (const float*)d_in[10];
    const float* ln_g       = (const float*)d_in[11];
    const float* ln_b       = (const float*)d_in[12];
    float* out = (float*)d_out;

    // workspace layout (floats): proj(8192x2144) | dt(8192x2048) | 4 small vecs
    float* P   = (float*)d_ws;
    float* DT  = P  + (size_t)NROWS * PCOLS;
    float* zl  = DT + (size_t)NROWS * D_INNER;
    float* yl  = zl  + B_SZ * D_INNER;
    float* xcl = yl  + B_SZ * D_INNER;
    float* fm  = xcl + B_SZ * D_INNER;

    gemm_inproj<<<dim3((PCOLS + 63) / 64, NROWS / 128), dim3(32, 4), 0, stream>>>(
        x, in_proj_w, in_proj_b, P);
    gemm_dt<<<dim3(D_INNER / 64, NROWS / 64), dim3(32, 4), 0, stream>>>(
        P, dt_proj_w, dt_proj_b, DT);
    zlast_kernel<<<(B_SZ * D_INNER) / 256, 256, 0, stream>>>(x, in_proj_w, in_proj_b, zl);
    scan_kernel<<<dim3(D_INNER / 16, B_SZ), 256, 0, stream>>>(
        P, DT, conv_w, conv_b, A_log, yl, xcl);
    finalize_ln<<<B_SZ, 256, 0, stream>>>(yl, xcl, zl, D_param, ln_g, ln_b, fm);
    outproj_kernel<<<(B_SZ * D_MODEL) / 256, 256, 0, stream>>>(fm, out_proj_w, out_proj_b, out);
}